// TAttn_73315091743306
// MI455X (gfx1250) — compile-verified
//
#include <hip/hip_runtime.h>
#include <hip/hip_bf16.h>
#include <math.h>

// ---- problem constants (match reference) ----
#define DIM_IN   16
#define DIM_OUT  64
#define BB       4
#define TT       8
#define NN       256

typedef __attribute__((ext_vector_type(2))) float v2f;
typedef __attribute__((ext_vector_type(8))) float v8f;

__device__ __forceinline__ float fast_tanh(float x) {
#if __has_builtin(__builtin_amdgcn_tanhf)
  return __builtin_amdgcn_tanhf(x);   // single v_tanh_f32 (TRANS) on gfx1250
#else
  return tanhf(x);
#endif
}

__device__ __forceinline__ float fast_exp(float x) {
#if __has_builtin(__builtin_amdgcn_exp2f)
  return __builtin_amdgcn_exp2f(x * 1.4426950408889634f); // v_exp_f32
#else
  return __expf(x);
#endif
}

// -----------------------------------------------------------------------------
// Projection: Y[rows,64] = X[rows,16] @ W[16,64] + bias[64]
// One wave per 16-row tile. Uses V_WMMA_F32_16X16X4_F32 with K chained 4x
// (K=16) over 4 N-tiles (N=64). Fragment layouts per CDNA5 ISA 7.12.2:
//   A (16x4 f32): lanes 0-15 -> K={kb,kb+1}, lanes 16-31 -> K={kb+2,kb+3},
//                 row M = lane%16.
//   B (4x16 f32): VGPR r holds row K=kb+2h+r, col N = n0 + lane%16 (h=lane/16).
//   C/D (16x16):  VGPR r holds (M = r + 8h, N = lane%16).
// -----------------------------------------------------------------------------
__global__ __launch_bounds__(256) void proj_wmma(
    const float* __restrict__ X, const float* __restrict__ W,
    const float* __restrict__ bias, float* __restrict__ Y, int rowTiles) {
  const int lane = threadIdx.x & 31;
  const int wave = threadIdx.x >> 5;
  const int tile = blockIdx.x * 8 + wave;
  if (tile >= rowTiles) return;          // wave-uniform; EXEC stays all-ones
  const int row0 = tile * 16;
  const int m16  = lane & 15;
  const int h    = lane >> 4;            // 0 or 1

  const float* xrow = X + (size_t)(row0 + m16) * DIM_IN;

  // A fragments for the 4 K-steps (K = 0..15 in chunks of 4)
  v2f a[4];
#pragma unroll
  for (int ks = 0; ks < 4; ++ks) {
    const int kb = ks * 4 + 2 * h;
    a[ks].x = xrow[kb + 0];
    a[ks].y = xrow[kb + 1];
  }

#pragma unroll
  for (int nt = 0; nt < 4; ++nt) {
    const int n0 = nt * 16;
    const float bcol = bias[n0 + m16];   // bias depends only on column
    v8f c;
#pragma unroll
    for (int r = 0; r < 8; ++r) c[r] = bcol;

#pragma unroll
    for (int ks = 0; ks < 4; ++ks) {
      const int kb = ks * 4;
      v2f b;
      b.x = W[(kb + 2 * h + 0) * DIM_OUT + n0 + m16];
      b.y = W[(kb + 2 * h + 1) * DIM_OUT + n0 + m16];
      // (neg_a, A, neg_b, B, c_mod, C, reuse_a, reuse_b)
      c = __builtin_amdgcn_wmma_f32_16x16x4_f32(false, a[ks], false, b,
                                                (short)0, c, false, false);
    }

#pragma unroll
    for (int r = 0; r < 8; ++r)
      Y[(size_t)(row0 + r + 8 * h) * DIM_OUT + n0 + m16] = c[r];
  }
}

// -----------------------------------------------------------------------------
// scores + softmax: one 256-thread block per (b,t,n); thread m owns column m.
//   s_m = sum_d wt[d] * tanh(q[b,n,d] + k[b,t,m,d]);   out = softmax_m(s)
// Trans-pipe bound: 64 v_tanh_f32 + 64 FMA per thread; q/wt staged in LDS,
// k rows stream from L2 (k_proj is 2MB, fully L2-resident on 192MB L2).
// -----------------------------------------------------------------------------
__global__ __launch_bounds__(256) void score_softmax(
    const float* __restrict__ qp,   // [B*N, 64]
    const float* __restrict__ kp,   // [B*T*N, 64]
    const float* __restrict__ wt,   // [64]
    float* __restrict__ out) {      // [B,T,N,N]
  __shared__ float qs[DIM_OUT];
  __shared__ float ws[DIM_OUT];
  __shared__ float red[256];

  const int idx = blockIdx.x;        // (b*T + t)*N + n
  const int n   = idx & (NN - 1);
  const int bt  = idx >> 8;          // b*T + t
  const int b   = bt >> 3;           // T == 8
  const int m   = threadIdx.x;

  if (m < DIM_OUT) {
    qs[m] = qp[(size_t)(b * NN + n) * DIM_OUT + m];
    ws[m] = wt[m];
  }
  __syncthreads();

  const float4* krow = (const float4*)(kp + (size_t)(bt * NN + m) * DIM_OUT);
  float acc = 0.f;
#pragma unroll
  for (int d4 = 0; d4 < DIM_OUT / 4; ++d4) {
    const float4 kv = krow[d4];
    const float4 qv = *(const float4*)(qs + 4 * d4);
    const float4 wv = *(const float4*)(ws + 4 * d4);
    acc += wv.x * fast_tanh(qv.x + kv.x);
    acc += wv.y * fast_tanh(qv.y + kv.y);
    acc += wv.z * fast_tanh(qv.z + kv.z);
    acc += wv.w * fast_tanh(qv.w + kv.w);
  }

  // ---- block softmax over 256 columns ----
  red[m] = acc;
  __syncthreads();
#pragma unroll
  for (int s = 128; s > 0; s >>= 1) {
    if (m < s) red[m] = fmaxf(red[m], red[m + s]);
    __syncthreads();
  }
  const float mx = red[0];
  __syncthreads();

  const float e = fast_exp(acc - mx);
  red[m] = e;
  __syncthreads();
#pragma unroll
  for (int s = 128; s > 0; s >>= 1) {
    if (m < s) red[m] += red[m + s];
    __syncthreads();
  }
  const float inv = 1.f / red[0];

  out[(size_t)idx * NN + m] = e * inv;
}

// -----------------------------------------------------------------------------
extern "C" void kernel_launch(void* const* d_in, const int* in_sizes, int n_in,
                              void* d_out, int out_size, void* d_ws,
                              size_t ws_size, hipStream_t stream) {
  const float* query = (const float*)d_in[0];  // [B,N,16]
  const float* keys  = (const float*)d_in[1];  // [B,T,N,16]
  const float* Wq    = (const float*)d_in[2];  // [16,64]
  const float* bq    = (const float*)d_in[3];  // [64]
  const float* Wk    = (const float*)d_in[4];  // [16,64]
  const float* bk    = (const float*)d_in[5];  // [64]
  const float* wt    = (const float*)d_in[6];  // [64]
  float* out = (float*)d_out;                  // [B,T,N,N]

  float* qp = (float*)d_ws;                          // [B*N, 64]   256 KB
  float* kp = qp + (size_t)BB * NN * DIM_OUT;        // [B*T*N, 64] 2 MB

  const int qTiles = (BB * NN) / 16;            // 64
  const int kTiles = (BB * TT * NN) / 16;       // 512

  proj_wmma<<<dim3((qTiles + 7) / 8), 256, 0, stream>>>(query, Wq, bq, qp, qTiles);
  proj_wmma<<<dim3((kTiles + 7) / 8), 256, 0, stream>>>(keys, Wk, bk, kp, kTiles);
  score_softmax<<<dim3(BB * TT * NN), 256, 0, stream>>>(qp, kp, wt, out);
}